// HCF_48232482734601
// MI455X (gfx1250) — compile-verified
//
#include <hip/hip_runtime.h>

// ---------------------------------------------------------------------------
// LightGCN-style propagation over 4 bipartite graphs (COO SpMM, d = 128 fp32).
// Memory-bound scatter/gather: per edge, gather a 512B row, scale, atomic-add
// a 512B row. One wave32 per edge: lane j handles floats [4j, 4j+4).
// Edge triples are staged per block into LDS via CDNA5 async-to-LDS loads
// (guarded; plain fallback), next gather row is prefetched (global_prefetch_b8).
// ---------------------------------------------------------------------------

#define TILE 256       // edges staged per block (8 waves x 32 edges)
#define EW_BLOCK 256

typedef __attribute__((address_space(1))) int g_int;   // global int
typedef __attribute__((address_space(3))) int l_int;   // LDS int

#if defined(__gfx1250__) && __has_builtin(__builtin_amdgcn_global_load_async_to_lds_b32)
#define USE_ASYNC_LDS 1
#else
#define USE_ASYNC_LDS 0
#endif

#if USE_ASYNC_LDS
__device__ __forceinline__ void async_b32(const void* g, void* l) {
  // global -> LDS async copy, 4 bytes per enabled lane (ASYNCcnt-tracked).
  // Builtin signature takes addrspace(1) int* and addrspace(3) int*;
  // go through integer casts to produce the address-space-qualified pointers.
  __builtin_amdgcn_global_load_async_to_lds_b32(
      (g_int*)(unsigned long long)g,
      (l_int*)(unsigned)(unsigned long long)l,
      0, 0);
}
__device__ __forceinline__ void wait_async0() {
#if __has_builtin(__builtin_amdgcn_s_wait_asynccnt)
  __builtin_amdgcn_s_wait_asynccnt(0);
#else
  asm volatile("s_wait_asynccnt 0" ::: "memory");
#endif
}
#endif

// y[oidx[e]] += vals[e] * x[iidx[e]]   for all edges e, rows of 128 floats.
__global__ void __launch_bounds__(TILE) spmm_kernel(
    const float* __restrict__ x, float* __restrict__ y,
    const int* __restrict__ oidx, const int* __restrict__ iidx,
    const float* __restrict__ vals, int nnz) {
  __shared__ int   s_o[TILE];
  __shared__ int   s_i[TILE];
  __shared__ float s_v[TILE];

  const int t    = threadIdx.x;
  const int base = blockIdx.x * TILE;
  const int g    = base + t;

#if USE_ASYNC_LDS
  if (g < nnz) {
    async_b32(oidx + g, &s_o[t]);
    async_b32(iidx + g, &s_i[t]);
    async_b32(vals + g, &s_v[t]);
  }
  wait_async0();        // each wave drains its own ASYNCcnt
  __syncthreads();      // then all 256 staged edges are visible block-wide
#else
  if (g < nnz) {
    s_o[t] = oidx[g];
    s_i[t] = iidx[g];
    s_v[t] = vals[g];
  }
  __syncthreads();
#endif

  const int cnt  = min(TILE, nnz - base);
  const int wave = t >> 5;
  const int lane = t & 31;
  const int jEnd = min((wave + 1) * 32, cnt);

  for (int j = wave * 32; j < jEnd; ++j) {
    const int   oi = s_o[j];
    const int   ii = s_i[j];
    const float v  = s_v[j];
    if (j + 1 < jEnd) {
      // prefetch next edge's source row (emits global_prefetch_b8 on gfx1250)
      __builtin_prefetch(x + (size_t)s_i[j + 1] * 128 + lane * 4, 0, 3);
    }
    // 32 lanes x float4 = full 128-float row -> global_load_b128 per lane
    const float4 p = ((const float4*)(x + (size_t)ii * 128))[lane];
    float* dst = y + (size_t)oi * 128 + (size_t)lane * 4;
    atomicAdd(dst + 0, v * p.x);   // global_atomic_add_f32, no return
    atomicAdd(dst + 1, v * p.y);
    atomicAdd(dst + 2, v * p.z);
    atomicAdd(dst + 3, v * p.w);
  }
}

// a = src; b = src
__global__ void k_copy2(const float4* __restrict__ src, float4* __restrict__ a,
                        float4* __restrict__ b, int n4) {
  int i = blockIdx.x * blockDim.x + threadIdx.x;
  if (i < n4) {
    float4 v = src[i];
    a[i] = v;
    b[i] = v;
  }
}

__global__ void k_zero(float4* __restrict__ p, int n4) {
  int i = blockIdx.x * blockDim.x + threadIdx.x;
  if (i < n4) {
    float4 z; z.x = 0.f; z.y = 0.f; z.z = 0.f; z.w = 0.f;
    p[i] = z;
  }
}

__global__ void k_add(float4* __restrict__ acc, const float4* __restrict__ v, int n4) {
  int i = blockIdx.x * blockDim.x + threadIdx.x;
  if (i < n4) {
    float4 a = acc[i];
    float4 b = v[i];
    a.x += b.x; a.y += b.y; a.z += b.z; a.w += b.w;
    acc[i] = a;
  }
}

__global__ void k_scale(float4* __restrict__ p, float s, int n4) {
  int i = blockIdx.x * blockDim.x + threadIdx.x;
  if (i < n4) {
    float4 a = p[i];
    a.x *= s; a.y *= s; a.z *= s; a.w *= s;
    p[i] = a;
  }
}

extern "C" void kernel_launch(void* const* d_in, const int* in_sizes, int n_in,
                              void* d_out, int out_size, void* d_ws, size_t ws_size,
                              hipStream_t stream) {
  (void)n_in; (void)out_size; (void)ws_size;

  const float* emb[4];
  const float* vals[4];
  const int*   rows[4];
  const int*   cols[4];
  int          nnz[4];
  for (int k = 0; k < 4; ++k) {
    emb[k]  = (const float*)d_in[k];
    vals[k] = (const float*)d_in[4 + k];
    rows[k] = (const int*)d_in[8 + 2 * k];
    cols[k] = (const int*)d_in[9 + 2 * k];
    nnz[k]  = in_sizes[4 + k];
  }

  static const int    Nn[4]     = {100000, 40000, 20000, 10000};
  static const int    Pp[4]     = {40000, 100000, 100000, 40000};
  static const size_t rowOff[4] = {0, 100000, 140000, 160000};

  float* outp = (float*)d_out;
  // workspace: cur (<= 100000x128 f32) then partner buffer t (<= 100000x128 f32)
  float* cur = (float*)d_ws;
  float* tb  = cur + (size_t)100000 * 128;

  for (int gI = 0; gI < 4; ++gI) {
    const size_t nE = (size_t)Nn[gI] * 128;
    const size_t pE = (size_t)Pp[gI] * 128;
    const int n4 = (int)(nE / 4);
    const int p4 = (int)(pE / 4);
    float* acc = outp + rowOff[gI] * 128;

    // acc = emb ; cur = emb
    k_copy2<<<(n4 + EW_BLOCK - 1) / EW_BLOCK, EW_BLOCK, 0, stream>>>(
        (const float4*)emb[gI], (float4*)acc, (float4*)cur, n4);

    const int eBlocks = (nnz[gI] + TILE - 1) / TILE;
    for (int L = 0; L < 3; ++L) {
      // t = A * cur     (t[rows[e]] += v[e] * cur[cols[e]])
      k_zero<<<(p4 + EW_BLOCK - 1) / EW_BLOCK, EW_BLOCK, 0, stream>>>((float4*)tb, p4);
      spmm_kernel<<<eBlocks, TILE, 0, stream>>>(cur, tb, rows[gI], cols[gI], vals[gI], nnz[gI]);

      // cur = A^T * t   (cur[cols[e]] += v[e] * t[rows[e]])
      k_zero<<<(n4 + EW_BLOCK - 1) / EW_BLOCK, EW_BLOCK, 0, stream>>>((float4*)cur, n4);
      spmm_kernel<<<eBlocks, TILE, 0, stream>>>(tb, cur, cols[gI], rows[gI], vals[gI], nnz[gI]);

      // acc += cur
      k_add<<<(n4 + EW_BLOCK - 1) / EW_BLOCK, EW_BLOCK, 0, stream>>>(
          (float4*)acc, (const float4*)cur, n4);
    }

    // mean over [emb, layer1..3]
    k_scale<<<(n4 + EW_BLOCK - 1) / EW_BLOCK, EW_BLOCK, 0, stream>>>(
        (float4*)acc, 0.25f, n4);
  }
}